// GCNBlock_66383014527704
// MI455X (gfx1250) — compile-verified
//
#include <hip/hip_runtime.h>
#include <math.h>

// CDNA5 wave32 WMMA types
typedef float v2f __attribute__((ext_vector_type(2)));
typedef float v8f __attribute__((ext_vector_type(8)));

#define BN_EPS 1e-5f

// ---------------------------------------------------------------------------
// init: deg[i] = 1.0 (self loop), zero BN stats accumulators
// ---------------------------------------------------------------------------
__global__ void init_kernel(float* __restrict__ deg, float* __restrict__ stats, int N) {
    int i = blockIdx.x * blockDim.x + threadIdx.x;
    if (i < 256) stats[i] = 0.0f;
    if (i < N)   deg[i]  = 1.0f;
}

// ---------------------------------------------------------------------------
// degree: deg[col[e]] += 1  (targets)
// ---------------------------------------------------------------------------
__global__ void degree_kernel(const int* __restrict__ col, float* __restrict__ deg, int E) {
    int e = blockIdx.x * blockDim.x + threadIdx.x;
    if (e < E) atomicAdd(&deg[col[e]], 1.0f);
}

// ---------------------------------------------------------------------------
// dinv = rsqrt(deg) in place (deg >= 1 always due to self loops)
// ---------------------------------------------------------------------------
__global__ void rsqrt_kernel(float* __restrict__ deg, int N) {
    int i = blockIdx.x * blockDim.x + threadIdx.x;
    if (i < N) deg[i] = __frsqrt_rn(deg[i]);
}

// ---------------------------------------------------------------------------
// GEMM h = x @ W via V_WMMA_F32_16X16X4_F32 (exact fp32), fused epilogue:
//   h[m][n]   stored for edge gather
//   out[m][n] = h[m][n] * dinv[m]^2   (self-loop contribution, init of accum)
// One wave handles a 16-row strip x all 128 cols: 8 v8f accumulators,
// 32 k-steps x 8 n-tiles = 256 wmma per wave.
// W (64KB) staged in LDS pre-swizzled into WMMA B layout:
//   ldsW[ktile][ntile][lane] = { W[4*kt + 2*(lane/16)    ][16*nt + lane%16],
//                                W[4*kt + 2*(lane/16) + 1][16*nt + lane%16] }
// ---------------------------------------------------------------------------
__global__ __launch_bounds__(256) void gemm_wmma_kernel(
    const float* __restrict__ x, const float* __restrict__ W,
    const float* __restrict__ dinv, float* __restrict__ h,
    float* __restrict__ out, int N)
{
    __shared__ float2 ldsW[32 * 8 * 32];  // 64 KB

    const int tid = threadIdx.x;
    // cooperative swizzled fill of W into LDS
    for (int idx = tid; idx < 32 * 8 * 32; idx += 256) {
        int lane  = idx & 31;
        int nt    = (idx >> 5) & 7;
        int kt    = idx >> 8;
        int k0    = 4 * kt + 2 * (lane >> 4);
        int n     = 16 * nt + (lane & 15);
        float2 w2;
        w2.x = W[(size_t)k0 * 128 + n];
        w2.y = W[(size_t)(k0 + 1) * 128 + n];
        ldsW[idx] = w2;
    }
    __syncthreads();

    const int lane  = tid & 31;
    const int strip = blockIdx.x * 8 + (tid >> 5);   // 16-row strip per wave
    if (strip * 16 >= N) return;                      // wave-uniform

    // A operand: lane holds row m = lane%16, K pair base 2*(lane/16)
    const float* xrow = x + (size_t)(strip * 16 + (lane & 15)) * 128 + 2 * (lane >> 4);

    v8f acc[8] = {};
    for (int kt = 0; kt < 32; ++kt) {
        float2 af = *(const float2*)(xrow + 4 * kt);
        v2f a; a.x = af.x; a.y = af.y;
        const float2* wrow = &ldsW[kt * 256 + lane];
#pragma unroll
        for (int nt = 0; nt < 8; ++nt) {
            float2 bf = wrow[nt * 32];
            v2f b; b.x = bf.x; b.y = bf.y;
            // (neg_a, A, neg_b, B, c_mod, C, reuse_a, reuse_b)
            acc[nt] = __builtin_amdgcn_wmma_f32_16x16x4_f32(
                false, a, false, b, (short)0, acc[nt], false, false);
        }
    }

    // Epilogue: D layout -> row m = strip*16 + r + 8*(lane/16), col = nt*16 + lane%16
    const int m_base = strip * 16 + ((lane >> 4) << 3);
    const int cn     = lane & 15;
#pragma unroll
    for (int r = 0; r < 8; ++r) {
        const int m   = m_base + r;
        const float d = dinv[m];
        const float d2 = d * d;
        size_t rowoff = (size_t)m * 128 + cn;
#pragma unroll
        for (int nt = 0; nt < 8; ++nt) {
            float v = acc[nt][r];
            h[rowoff + nt * 16]   = v;
            out[rowoff + nt * 16] = v * d2;
        }
    }
}

// ---------------------------------------------------------------------------
// Edge scatter: one wave per edge. Lane l handles 4 consecutive features.
//   out[col] += h[row] * (dinv[row]*dinv[col])
// ---------------------------------------------------------------------------
__global__ __launch_bounds__(256) void edge_kernel(
    const int* __restrict__ row, const int* __restrict__ col,
    const float* __restrict__ dinv, const float* __restrict__ h,
    float* __restrict__ out, int E)
{
    int gid  = blockIdx.x * blockDim.x + threadIdx.x;
    int e    = gid >> 5;
    int lane = gid & 31;
    if (e >= E) return;                               // wave-uniform
    int r = row[e];
    int c = col[e];
    float norm = dinv[r] * dinv[c];
    float4 hv = ((const float4*)(h + (size_t)r * 128))[lane];
    float* dst = out + (size_t)c * 128 + lane * 4;
    atomicAdd(dst + 0, hv.x * norm);
    atomicAdd(dst + 1, hv.y * norm);
    atomicAdd(dst + 2, hv.z * norm);
    atomicAdd(dst + 3, hv.w * norm);
}

// ---------------------------------------------------------------------------
// BN stats: per-column sum and sum-of-squares (bias b cancels inside BN)
// ---------------------------------------------------------------------------
__global__ __launch_bounds__(128) void stats_kernel(
    const float* __restrict__ out, float* __restrict__ stats, int N)
{
    int f = threadIdx.x;                // 128 features
    float s = 0.0f, s2 = 0.0f;
    for (int i = blockIdx.x; i < N; i += gridDim.x) {
        float v = out[(size_t)i * 128 + f];
        s  += v;
        s2 += v * v;
    }
    atomicAdd(&stats[f],       s);
    atomicAdd(&stats[128 + f], s2);
}

// ---------------------------------------------------------------------------
// BN (training-mode batch stats) + ReLU, in place on d_out
// ---------------------------------------------------------------------------
__global__ __launch_bounds__(256) void bn_relu_kernel(
    const float* __restrict__ stats, const float* __restrict__ gamma,
    const float* __restrict__ beta, float* __restrict__ out, int N)
{
    size_t total = (size_t)N * 128;
    size_t stride = (size_t)gridDim.x * blockDim.x;
    for (size_t idx = (size_t)blockIdx.x * blockDim.x + threadIdx.x;
         idx < total; idx += stride) {
        int f = (int)(idx & 127);
        float invN = 1.0f / (float)N;
        float mean = stats[f] * invN;
        float var  = stats[128 + f] * invN - mean * mean;
        float inv  = __frsqrt_rn(var + BN_EPS);
        float y = (out[idx] - mean) * inv * gamma[f] + beta[f];
        out[idx] = fmaxf(y, 0.0f);
    }
}

// ---------------------------------------------------------------------------
extern "C" void kernel_launch(void* const* d_in, const int* in_sizes, int n_in,
                              void* d_out, int out_size, void* d_ws, size_t ws_size,
                              hipStream_t stream) {
    const float* x     = (const float*)d_in[0];
    const int*   ei    = (const int*)d_in[1];
    const float* W     = (const float*)d_in[2];
    // d_in[3] = b : cancels exactly inside BatchNorm, unused
    const float* gamma = (const float*)d_in[4];
    const float* beta  = (const float*)d_in[5];

    const int N = in_sizes[0] / 128;
    const int E = in_sizes[1] / 2;
    const int* row = ei;        // sources
    const int* col = ei + E;    // targets

    // workspace layout
    char* ws = (char*)d_ws;
    float* deg   = (float*)ws;                                 // N floats (becomes dinv)
    size_t off_h = (((size_t)N * 4 + 255) / 256) * 256;
    float* h     = (float*)(ws + off_h);                       // N*128 floats
    float* stats = (float*)(ws + off_h + (size_t)N * 128 * 4); // 256 floats

    float* out = (float*)d_out;                                // accumulator, then final y

    // 1) deg init + stats zero
    init_kernel<<<(N + 255) / 256, 256, 0, stream>>>(deg, stats, N);
    // 2) degree via atomics on targets
    degree_kernel<<<(E + 255) / 256, 256, 0, stream>>>(col, deg, E);
    // 3) dinv = rsqrt(deg)
    rsqrt_kernel<<<(N + 255) / 256, 256, 0, stream>>>(deg, N);
    // 4) WMMA GEMM + self-loop epilogue (writes every element of out)
    int strips = (N + 15) / 16;
    gemm_wmma_kernel<<<(strips + 7) / 8, 256, 0, stream>>>(x, W, deg, h, out, N);
    // 5) edge scatter (one wave per edge)
    edge_kernel<<<((size_t)E * 32 + 255) / 256, 256, 0, stream>>>(row, col, deg, h, out, E);
    // 6) BN column stats
    stats_kernel<<<512, 128, 0, stream>>>(out, stats, N);
    // 7) BN + ReLU in place
    bn_relu_kernel<<<4096, 256, 0, stream>>>(stats, gamma, beta, out, N);
}